// SentenceFeatureNormalization_90787018703651
// MI455X (gfx1250) — compile-verified
//
#include <hip/hip_runtime.h>
#include <hip/hip_bf16.h>

typedef float v2f __attribute__((ext_vector_type(2)));
typedef float v8f __attribute__((ext_vector_type(8)));
typedef float f4  __attribute__((ext_vector_type(4)));

#define B_  32
#define T_  8192
#define F_  256
#define CH_ 64
#define TC_ (T_ / CH_)   /* 128 rows per chunk */

// ---------------------------------------------------------------------------
// Kernel 1: masked partial sum / sum-of-squares per (b, chunk, f) using
// V_WMMA_F32_16X16X4_F32 with a ones A-matrix as a column-sum engine.
//
// Wave covers 64 features via 4 WMMA tiles; tile j, column n holds feature
// fbase + 4n + j, so one 4-row step = 2 coalesced global_load_b128 feeding
// 8 WMMAs, and extraction packs into 2 b128 stores. Chunk classification is
// wave-uniform: fully-masked -> zero partials; fully-valid -> maskless fast
// loop; boundary (at most one per batch row) -> per-row vector selects.
// EXEC stays all-1s at every WMMA. 8-wave block processes 2 chunks.
// ---------------------------------------------------------------------------
__global__ __launch_bounds__(256) void sfn_stats_partial(
    const float* __restrict__ x, const int* __restrict__ lengths,
    float* __restrict__ part /* [B][CH][2][F] */)
{
  const int blk   = blockIdx.x;
  const int b     = blk / (CH_ / 2);
  const int cpair = blk % (CH_ / 2);

  const int lane = threadIdx.x & 31;
  const int wave = threadIdx.x >> 5;
  const int g    = wave >> 2;          // chunk within pair (0/1)
  const int wf   = wave & 3;           // feature group (64 features each)

  const int c     = cpair * 2 + g;
  const int t0    = c * TC_;
  const int fbase = wf * 64;
  const int n     = lane & 15;
  const int kh    = lane >> 4;
  const int len   = lengths[b];

  float* outp = part + (((size_t)b * CH_ + c) * 2) * F_;

  if (t0 >= len) {  // wave-uniform: fully masked chunk -> zero partials
    if (lane < 16) {
      f4 z = {0.0f, 0.0f, 0.0f, 0.0f};
      *(f4*)(outp + fbase + 4 * lane)      = z;
      *(f4*)(outp + F_ + fbase + 4 * lane) = z;
    }
    return;
  }

  v2f ones; ones.x = 1.0f; ones.y = 1.0f;
  v8f aS0 = {}; v8f aS1 = {}; v8f aS2 = {}; v8f aS3 = {};
  v8f aQ0 = {}; v8f aQ1 = {}; v8f aQ2 = {}; v8f aQ3 = {};

  const f4 z4 = {0.0f, 0.0f, 0.0f, 0.0f};
  // base points at (row t0 + 2*kh, feature fbase + 4n)
  const float* base =
      x + ((size_t)b * T_ + t0 + 2 * kh) * F_ + fbase + 4 * n;

  if (t0 + TC_ <= len) {
    // ---- fast path: chunk fully valid, no masking -------------------------
    for (int t = 0; t < TC_; t += 4) {
      const float* p = base + (size_t)t * F_;
      f4 A  = *(const f4*)(p);        // row tr0,   4 features
      f4 Bv = *(const f4*)(p + F_);   // row tr0+1, 4 features

      v2f v0; v0.x = A.x; v0.y = Bv.x;
      v2f v1; v1.x = A.y; v1.y = Bv.y;
      v2f v2; v2.x = A.z; v2.y = Bv.z;
      v2f v3; v3.x = A.w; v3.y = Bv.w;
      v2f q0 = v0 * v0;
      v2f q1 = v1 * v1;
      v2f q2 = v2 * v2;
      v2f q3 = v3 * v3;

      aS0 = __builtin_amdgcn_wmma_f32_16x16x4_f32(false, ones, false, v0, (short)0, aS0, false, false);
      aS1 = __builtin_amdgcn_wmma_f32_16x16x4_f32(false, ones, false, v1, (short)0, aS1, false, false);
      aS2 = __builtin_amdgcn_wmma_f32_16x16x4_f32(false, ones, false, v2, (short)0, aS2, false, false);
      aS3 = __builtin_amdgcn_wmma_f32_16x16x4_f32(false, ones, false, v3, (short)0, aS3, false, false);
      aQ0 = __builtin_amdgcn_wmma_f32_16x16x4_f32(false, ones, false, q0, (short)0, aQ0, false, false);
      aQ1 = __builtin_amdgcn_wmma_f32_16x16x4_f32(false, ones, false, q1, (short)0, aQ1, false, false);
      aQ2 = __builtin_amdgcn_wmma_f32_16x16x4_f32(false, ones, false, q2, (short)0, aQ2, false, false);
      aQ3 = __builtin_amdgcn_wmma_f32_16x16x4_f32(false, ones, false, q3, (short)0, aQ3, false, false);
    }
  } else {
    // ---- boundary chunk (at most one per batch row): per-row selects ------
    for (int t = 0; t < TC_; t += 4) {
      const int rowA = t0 + t + 2 * kh;     // component-0 row
      const int rowB = rowA + 1;            // component-1 row
      const float* p = base + (size_t)t * F_;

      f4 A  = *(const f4*)(p);
      f4 Bv = *(const f4*)(p + F_);
      A  = (rowA < len) ? A  : z4;          // vector selects (no branch)
      Bv = (rowB < len) ? Bv : z4;

      v2f v0; v0.x = A.x; v0.y = Bv.x;
      v2f v1; v1.x = A.y; v1.y = Bv.y;
      v2f v2; v2.x = A.z; v2.y = Bv.z;
      v2f v3; v3.x = A.w; v3.y = Bv.w;
      v2f q0 = v0 * v0;
      v2f q1 = v1 * v1;
      v2f q2 = v2 * v2;
      v2f q3 = v3 * v3;

      aS0 = __builtin_amdgcn_wmma_f32_16x16x4_f32(false, ones, false, v0, (short)0, aS0, false, false);
      aS1 = __builtin_amdgcn_wmma_f32_16x16x4_f32(false, ones, false, v1, (short)0, aS1, false, false);
      aS2 = __builtin_amdgcn_wmma_f32_16x16x4_f32(false, ones, false, v2, (short)0, aS2, false, false);
      aS3 = __builtin_amdgcn_wmma_f32_16x16x4_f32(false, ones, false, v3, (short)0, aS3, false, false);
      aQ0 = __builtin_amdgcn_wmma_f32_16x16x4_f32(false, ones, false, q0, (short)0, aQ0, false, false);
      aQ1 = __builtin_amdgcn_wmma_f32_16x16x4_f32(false, ones, false, q1, (short)0, aQ1, false, false);
      aQ2 = __builtin_amdgcn_wmma_f32_16x16x4_f32(false, ones, false, q2, (short)0, aQ2, false, false);
      aQ3 = __builtin_amdgcn_wmma_f32_16x16x4_f32(false, ones, false, q3, (short)0, aQ3, false, false);
    }
  }

  // D layout: VGPR0, lanes 0..15 = (M=0, N=lane). Tile j column l holds
  // feature fbase + 4l + j -> lane l owns 4 consecutive features: b128 store.
  if (lane < 16) {
    f4 sv; sv.x = aS0[0]; sv.y = aS1[0]; sv.z = aS2[0]; sv.w = aS3[0];
    f4 qv; qv.x = aQ0[0]; qv.y = aQ1[0]; qv.z = aQ2[0]; qv.w = aQ3[0];
    *(f4*)(outp + fbase + 4 * lane)      = sv;
    *(f4*)(outp + F_ + fbase + 4 * lane) = qv;
  }
}

// ---------------------------------------------------------------------------
// Kernel 2: reduce chunk partials -> mean, rstd per (b, f)
// var = (sum_sq - sum*mean) / max(n-1, 1)   (identity of masked Σ(x-mean)^2)
// ---------------------------------------------------------------------------
__global__ __launch_bounds__(256) void sfn_stats_final(
    const float* __restrict__ part, const int* __restrict__ lengths,
    float* __restrict__ stats /* [B][2][F] */)
{
  const int b = blockIdx.x;
  const int f = threadIdx.x;

  float s = 0.0f, q = 0.0f;
  const float* p = part + ((size_t)b * CH_ * 2) * F_ + f;
  for (int c = 0; c < CH_; ++c) {
    s += p[(size_t)(2 * c) * F_];
    q += p[(size_t)(2 * c + 1) * F_];
  }

  const float nf   = (float)lengths[b];
  const float mean = s / nf;
  float var = (q - s * mean) / fmaxf(nf - 1.0f, 1.0f);
  var = fmaxf(var, 0.0f);
  const float stdv = fmaxf(sqrtf(var), 1e-10f);

  stats[((size_t)b * 2) * F_ + f]     = mean;
  stats[((size_t)b * 2 + 1) * F_ + f] = 1.0f / stdv;
}

// ---------------------------------------------------------------------------
// Kernel 3: out = (x - mean) * rstd over the FULL sequence (incl. padding).
// NT load/store for the streamed tensor (touched exactly once here); x may
// still hit L2 from kernel 1's RT fills. Stats reads stay cached.
// ---------------------------------------------------------------------------
__global__ __launch_bounds__(256) void sfn_normalize(
    const float* __restrict__ x, const float* __restrict__ stats,
    float* __restrict__ out)
{
  const size_t gid = (size_t)blockIdx.x * blockDim.x + threadIdx.x;  // B*T*(F/4)
  const int    fq  = (int)(gid & 63);        // F/4 = 64
  const size_t bt  = gid >> 6;
  const int    b   = (int)(bt >> 13);        // T = 8192

  const f4* xv = (const f4*)x;
  f4 v = __builtin_nontemporal_load(xv + gid);

  const f4* mv = (const f4*)(stats + (size_t)b * 2 * F_);
  const f4* rv = (const f4*)(stats + (size_t)b * 2 * F_ + F_);
  f4 m = mv[fq];
  f4 r = rv[fq];

  f4 o = (v - m) * r;
  __builtin_nontemporal_store(o, (f4*)out + gid);
}

// ---------------------------------------------------------------------------
extern "C" void kernel_launch(void* const* d_in, const int* in_sizes, int n_in,
                              void* d_out, int out_size, void* d_ws, size_t ws_size,
                              hipStream_t stream)
{
  (void)in_sizes; (void)n_in; (void)out_size; (void)ws_size;

  const float* x       = (const float*)d_in[0];
  const int*   lengths = (const int*)d_in[1];
  float*       out     = (float*)d_out;

  float* part  = (float*)d_ws;                        // B*CH*2*F floats (4 MiB)
  float* stats = part + (size_t)B_ * CH_ * 2 * F_;    // B*2*F floats

  sfn_stats_partial<<<B_ * (CH_ / 2), 256, 0, stream>>>(x, lengths, part);
  sfn_stats_final<<<B_, 256, 0, stream>>>(part, lengths, stats);
  sfn_normalize<<<(B_ * T_ * (F_ / 4)) / 256, 256, 0, stream>>>(x, stats, out);
}